// QuaternionBatchNorm_38663295598901
// MI455X (gfx1250) — compile-verified
//
#include <hip/hip_runtime.h>
#include <math.h>

#define QF 256
#define EPSQ 1e-5f

typedef float v2f __attribute__((ext_vector_type(2)));
typedef float v8f __attribute__((ext_vector_type(8)));

// ---------------------------------------------------------------------------
// Pass 1: per-block partial sums of the 14 statistics per feature
//   k=0..3  : sum x_a          k=4..13 : sum x_a*x_b (upper-tri order
//   00,01,02,03,11,12,13,22,23,33)
// Layout of partials: part[(f*14+k)*NB + b]  (reducer reads contiguous runs)
// ---------------------------------------------------------------------------
__global__ void __launch_bounds__(1024) qbn_partial(
    const float* __restrict__ x, float* __restrict__ part,
    int N, int NB, int CH) {
  __shared__ float red[4 * 3584];
  const int tid  = threadIdx.x;
  const int f    = tid & 255;
  const int nsub = tid >> 8;          // 0..3
  const int b    = blockIdx.x;
  const long NF  = (long)N * QF;
  const int n0 = b * CH;
  const int n1 = min(N, n0 + CH);

  float s0=0.f,s1=0.f,s2=0.f,s3=0.f;
  float p00=0.f,p01=0.f,p02=0.f,p03=0.f,p11=0.f,p12=0.f,p13=0.f,
        p22=0.f,p23=0.f,p33=0.f;
  const float* xf = x + f;
#pragma unroll 2
  for (int n = n0 + nsub; n < n1; n += 4) {
    const long base = (long)n * QF;
    const float x0 = xf[base];
    const float x1 = xf[base + NF];
    const float x2 = xf[base + 2 * NF];
    const float x3 = xf[base + 3 * NF];
    s0 += x0; s1 += x1; s2 += x2; s3 += x3;
    p00 = fmaf(x0, x0, p00); p01 = fmaf(x0, x1, p01);
    p02 = fmaf(x0, x2, p02); p03 = fmaf(x0, x3, p03);
    p11 = fmaf(x1, x1, p11); p12 = fmaf(x1, x2, p12);
    p13 = fmaf(x1, x3, p13); p22 = fmaf(x2, x2, p22);
    p23 = fmaf(x2, x3, p23); p33 = fmaf(x3, x3, p33);
  }
  const float vals[14] = {s0,s1,s2,s3,p00,p01,p02,p03,p11,p12,p13,p22,p23,p33};
  const int base14 = nsub * 3584 + f * 14;
#pragma unroll
  for (int k = 0; k < 14; ++k) red[base14 + k] = vals[k];
  __syncthreads();
  // combine the 4 n-sublanes, write one partial per (f,k) per block
  for (int p = tid; p < 3584; p += 1024) {
    const float v = red[p] + red[3584 + p] + red[2 * 3584 + p] + red[3 * 3584 + p];
    part[(long)p * NB + b] = v;
  }
}

// ---------------------------------------------------------------------------
// Pass 1b: deterministic serial reduction over blocks (contiguous float4 runs)
// ---------------------------------------------------------------------------
__global__ void qbn_reduce(const float* __restrict__ part,
                           float* __restrict__ stats, int NB) {
  const int p = blockIdx.x * 256 + threadIdx.x;
  if (p >= 3584) return;
  const float4* src = (const float4*)(part + (long)p * NB);
  float4 a = {0.f, 0.f, 0.f, 0.f};
#pragma unroll 4
  for (int i = 0; i < (NB >> 2); ++i) {
    const float4 v = src[i];
    a.x += v.x; a.y += v.y; a.z += v.z; a.w += v.w;
  }
  stats[p] = (a.x + a.y) + (a.z + a.w);
}

// ---------------------------------------------------------------------------
// Pass 2: per-feature mean/cov -> Cholesky -> L^-1 -> M = W @ L^-1 (WMMA),
// c = bias - M @ mean.  One block of 256 threads (8 waves).
// M_f = W_f @ G_f done 4 features per V_WMMA_F32_16X16X4_F32 (block-diagonal
// packing): A rows = (feat, W-row), B cols = (feat, G-col), K = contraction.
// Output coeffT[k*256 + f], k<16: M[k/4][k%4], k>=16: c[k-16].
// ---------------------------------------------------------------------------
__global__ void __launch_bounds__(256) qbn_coeff(
    const float* __restrict__ stats, const float* __restrict__ weight,
    const float* __restrict__ bias, float* __restrict__ coeffT, int N) {
  __shared__ float Wl[16 * QF];     // weight, same flat layout [(a*4+b)*256+f]
  __shared__ float Gl[QF * 16];     // L^-1 row-major per feature
  __shared__ float Ml[QF * 16];     // M row-major per feature
  __shared__ float meanl[QF * 4];
  const int tid = threadIdx.x;
  for (int i = tid; i < 16 * QF; i += 256) Wl[i] = weight[i];
  {
    const int f = tid;
    float st[14];
#pragma unroll
    for (int k = 0; k < 14; ++k) st[k] = stats[f * 14 + k];
    const float invn = 1.0f / (float)N;
    const float m0 = st[0]*invn, m1 = st[1]*invn, m2 = st[2]*invn, m3 = st[3]*invn;
    const float v00 = st[4]*invn  - m0*m0 + EPSQ;
    const float v01 = st[5]*invn  - m0*m1;
    const float v02 = st[6]*invn  - m0*m2;
    const float v03 = st[7]*invn  - m0*m3;
    const float v11 = st[8]*invn  - m1*m1 + EPSQ;
    const float v12 = st[9]*invn  - m1*m2;
    const float v13 = st[10]*invn - m1*m3;
    const float v22 = st[11]*invn - m2*m2 + EPSQ;
    const float v23 = st[12]*invn - m2*m3;
    const float v33 = st[13]*invn - m3*m3 + EPSQ;
    // 4x4 lower Cholesky
    const float L00 = sqrtf(v00);            const float r00 = 1.0f / L00;
    const float L10 = v01 * r00, L20 = v02 * r00, L30 = v03 * r00;
    const float L11 = sqrtf(v11 - L10*L10);  const float r11 = 1.0f / L11;
    const float L21 = (v12 - L20*L10) * r11;
    const float L31 = (v13 - L30*L10) * r11;
    const float L22 = sqrtf(v22 - L20*L20 - L21*L21);  const float r22 = 1.0f / L22;
    const float L32 = (v23 - L30*L20 - L31*L21) * r22;
    const float L33 = sqrtf(v33 - L30*L30 - L31*L31 - L32*L32);
    const float r33 = 1.0f / L33;
    // G = L^-1 (lower triangular)
    const float G00 = r00, G11 = r11, G22 = r22, G33 = r33;
    const float G10 = -r11 * (L10 * G00);
    const float G21 = -r22 * (L21 * G11);
    const float G32 = -r33 * (L32 * G22);
    const float G20 = -r22 * (L20 * G00 + L21 * G10);
    const float G31 = -r33 * (L31 * G11 + L32 * G21);
    const float G30 = -r33 * (L30 * G00 + L31 * G10 + L32 * G20);
    float* g = Gl + f * 16;
    g[0]=G00; g[1]=0.f;  g[2]=0.f;  g[3]=0.f;
    g[4]=G10; g[5]=G11;  g[6]=0.f;  g[7]=0.f;
    g[8]=G20; g[9]=G21;  g[10]=G22; g[11]=0.f;
    g[12]=G30; g[13]=G31; g[14]=G32; g[15]=G33;
    meanl[f*4+0]=m0; meanl[f*4+1]=m1; meanl[f*4+2]=m2; meanl[f*4+3]=m3;
  }
  __syncthreads();
#if defined(__gfx1250__) && __has_builtin(__builtin_amdgcn_wmma_f32_16x16x4_f32)
  {
    const int lane = tid & 31, wave = tid >> 5;
    const int r    = lane & 15;                 // A-row / B-col index 0..15
    const int k0   = (lane < 16) ? 0 : 2;       // K pair held by this lane
    const int feat = r >> 2;                    // feature within group of 4
    const int rc   = r & 3;                     // W-row (A) / G-col (B)
    const bool active = ((lane < 16) == (feat < 2));
    const int vbase = 4 * feat - ((lane < 16) ? 0 : 8);
#pragma unroll 1
    for (int gi = 0; gi < 8; ++gi) {
      const int g  = wave * 8 + gi;             // 64 groups over 8 waves
      const int fA = 4 * g + feat;
      v2f av, bv;
      av.x = Wl[(rc * 4 + k0) * QF + fA];       // A[r,k]   = W[rc, k, fA]
      av.y = Wl[(rc * 4 + k0 + 1) * QF + fA];
      bv.x = Gl[fA * 16 + k0 * 4 + rc];         // B[k,col] = G[k, rc, fA]
      bv.y = Gl[fA * 16 + (k0 + 1) * 4 + rc];
      v8f cz = {0.f,0.f,0.f,0.f,0.f,0.f,0.f,0.f};
      v8f d = __builtin_amdgcn_wmma_f32_16x16x4_f32(
          false, av, false, bv, (short)0, cz, false, false);
      // diagonal block (feat==feat) rows of D live in d[vbase..vbase+3]
      const float w0 = (vbase == 0) ? d[0] : d[4];
      const float w1 = (vbase == 0) ? d[1] : d[5];
      const float w2 = (vbase == 0) ? d[2] : d[6];
      const float w3 = (vbase == 0) ? d[3] : d[7];
      if (active) {
        Ml[fA * 16 + 0 * 4 + rc] = w0;
        Ml[fA * 16 + 1 * 4 + rc] = w1;
        Ml[fA * 16 + 2 * 4 + rc] = w2;
        Ml[fA * 16 + 3 * 4 + rc] = w3;
      }
    }
  }
#else
  {
    const int f = tid;
#pragma unroll
    for (int a = 0; a < 4; ++a)
#pragma unroll
      for (int c = 0; c < 4; ++c) {
        float acc = 0.f;
#pragma unroll
        for (int b2 = 0; b2 < 4; ++b2)
          acc += Wl[(a * 4 + b2) * QF + f] * Gl[f * 16 + b2 * 4 + c];
        Ml[f * 16 + a * 4 + c] = acc;
      }
  }
#endif
  __syncthreads();
  {
    const int f = tid;
#pragma unroll
    for (int a = 0; a < 4; ++a) {
      float cb = bias[a * QF + f];
#pragma unroll
      for (int b2 = 0; b2 < 4; ++b2) {
        const float m = Ml[f * 16 + a * 4 + b2];
        cb -= m * meanl[f * 4 + b2];
        coeffT[(a * 4 + b2) * QF + f] = m;
      }
      coeffT[(16 + a) * QF + f] = cb;
    }
  }
}

// ---------------------------------------------------------------------------
// Pass 3: out[a,n,f] = c[a,f] + sum_b M[a,b,f] * x[b,n,f]   (streaming)
// ---------------------------------------------------------------------------
__global__ void __launch_bounds__(256) qbn_apply(
    const float* __restrict__ x, const float* __restrict__ coeffT,
    float* __restrict__ out, int N, int CH) {
  const int f  = threadIdx.x;
  const int n0 = blockIdx.x * CH;
  const int n1 = min(N, n0 + CH);
  const long NF = (long)N * QF;
  float M[16], c[4];
#pragma unroll
  for (int k = 0; k < 16; ++k) M[k] = coeffT[k * QF + f];
#pragma unroll
  for (int a = 0; a < 4; ++a) c[a] = coeffT[(16 + a) * QF + f];
  const float* xf = x + f;
  float* of = out + f;
#pragma unroll 2
  for (int n = n0; n < n1; ++n) {
    const long base = (long)n * QF;
    const float x0 = xf[base];
    const float x1 = xf[base + NF];
    const float x2 = xf[base + 2 * NF];
    const float x3 = xf[base + 3 * NF];
    const float o0 = c[0] + M[0]*x0  + M[1]*x1  + M[2]*x2  + M[3]*x3;
    const float o1 = c[1] + M[4]*x0  + M[5]*x1  + M[6]*x2  + M[7]*x3;
    const float o2 = c[2] + M[8]*x0  + M[9]*x1  + M[10]*x2 + M[11]*x3;
    const float o3 = c[3] + M[12]*x0 + M[13]*x1 + M[14]*x2 + M[15]*x3;
    __builtin_nontemporal_store(o0, of + base);
    __builtin_nontemporal_store(o1, of + base + NF);
    __builtin_nontemporal_store(o2, of + base + 2 * NF);
    __builtin_nontemporal_store(o3, of + base + 3 * NF);
  }
}

extern "C" void kernel_launch(void* const* d_in, const int* in_sizes, int n_in,
                              void* d_out, int out_size, void* d_ws, size_t ws_size,
                              hipStream_t stream) {
  const float* x      = (const float*)d_in[0];
  const float* weight = (const float*)d_in[1];
  const float* bias   = (const float*)d_in[2];
  float* out = (float*)d_out;
  const int N = in_sizes[0] / (4 * QF);

  // workspace: partials [3584*NB] | stats [3584] | coeffT [20*256]
  int NB = 256;
  while (NB > 32) {
    const size_t need = ((size_t)3584 * NB + 3584 + 20 * QF) * sizeof(float);
    if (need <= ws_size) break;
    NB >>= 1;
  }
  float* part   = (float*)d_ws;
  float* stats  = part + (size_t)3584 * NB;
  float* coeffT = stats + 3584;

  int CH = (N + NB - 1) / NB;
  CH = (CH + 3) & ~3;                       // multiple of 4 for the nsub striping
  qbn_partial<<<NB, 1024, 0, stream>>>(x, part, N, NB, CH);
  qbn_reduce<<<14, 256, 0, stream>>>(part, stats, NB);
  qbn_coeff<<<1, 256, 0, stream>>>(stats, weight, bias, coeffT, N);
  const int NB3 = 512;
  const int CH3 = (N + NB3 - 1) / NB3;
  qbn_apply<<<NB3, 256, 0, stream>>>(x, coeffT, out, N, CH3);
}